// PopulationCoding_68994354643160
// MI455X (gfx1250) — compile-verified
//
#include <hip/hip_runtime.h>
#include <cstdint>

typedef __attribute__((ext_vector_type(2))) float v2f;
typedef __attribute__((ext_vector_type(8))) float v8f;

#define B_DIM   1024
#define IN_DIM  512
#define D_DIM   256
#define P_DIM   8
#define TOT     (D_DIM * P_DIM)   // 2048
#define NSTEPS  32
#define BETA    0.95f

static __device__ __forceinline__ v8f wmma_f32(v2f a, v2f b, v8f c) {
    return __builtin_amdgcn_wmma_f32_16x16x4_f32(
        /*neg_a=*/false, a, /*neg_b=*/false, b,
        /*c_mod=*/(short)0, c, /*reuse_a=*/false, /*reuse_b=*/false);
}

// ---------------------------------------------------------------------------
// Kernel 1: I = x @ W_proj + b_proj (fp32 WMMA).
// One wave computes a 16(M) x 64(N) tile: 4 independent accumulators break
// the WMMA->WMMA C-operand RAW chain and amortize A-fragment loads 4x.
// ---------------------------------------------------------------------------
__global__ __launch_bounds__(256) void pc_gemm_wmma(
    const float* __restrict__ x,      // [1024, 512]
    const float* __restrict__ W,      // [512, 2048]
    const float* __restrict__ bias,   // [2048]
    float* __restrict__ I)            // [1024, 2048] (workspace)
{
    const int wave = (blockIdx.x * blockDim.x + threadIdx.x) >> 5;
    const int lane = threadIdx.x & 31;
    const int ntiles = TOT / 64;                   // 32 N-supertiles
    const int m0 = (wave / ntiles) * 16;
    const int n0 = (wave % ntiles) * 64;
    const int half = lane >> 4;                    // 0: K={0,1}, 1: K={2,3}
    const int lm   = lane & 15;

    // A fragment: row (m0+lm), K pair starting at 2*half (consecutive -> b64)
    const float* arow  = x + (size_t)(m0 + lm) * IN_DIM + 2 * half;
    // B fragments: cols n0+lm, n0+16+lm, n0+32+lm, n0+48+lm; K rows 2*half(+1)
    const float* bbase = W + (size_t)(2 * half) * TOT + n0 + lm;

    v8f acc0 = {}, acc1 = {}, acc2 = {}, acc3 = {};
    for (int k = 0; k < IN_DIM; k += 4) {
        v2f a;
        a.x = arow[k];
        a.y = arow[k + 1];
        const float* bk = bbase + (size_t)k * TOT;
        v2f b0, b1, b2, b3;
        b0.x = bk[0];   b0.y = bk[TOT + 0];
        b1.x = bk[16];  b1.y = bk[TOT + 16];
        b2.x = bk[32];  b2.y = bk[TOT + 32];
        b3.x = bk[48];  b3.y = bk[TOT + 48];
        // 4 independent WMMAs per k-step (no C-chain between them)
        acc0 = wmma_f32(a, b0, acc0);
        acc1 = wmma_f32(a, b1, acc1);
        acc2 = wmma_f32(a, b2, acc2);
        acc3 = wmma_f32(a, b3, acc3);
    }

    const float bn0 = bias[n0 + lm];
    const float bn1 = bias[n0 + 16 + lm];
    const float bn2 = bias[n0 + 32 + lm];
    const float bn3 = bias[n0 + 48 + lm];
#pragma unroll
    for (int j = 0; j < 8; ++j) {
        // D layout: VGPR j -> M = j + 8*half, N = lm
        float* row = I + (size_t)(m0 + j + 8 * half) * TOT + n0 + lm;
        row[0]  = acc0[j] + bn0;
        row[16] = acc1[j] + bn1;
        row[32] = acc2[j] + bn2;
        row[48] = acc3[j] + bn3;
    }
}

// ---------------------------------------------------------------------------
// Kernel 2: fused LIF recurrence (bit-packed spikes) + rate + temporal conv
// One thread per (b, d). Spike trains live as 8 x uint32 bitmasks.
// ---------------------------------------------------------------------------
__global__ __launch_bounds__(256) void pc_fused(
    const float* __restrict__ I,          // [B, D, P] = [B, 2048]
    const float* __restrict__ thresholds, // [8]
    const float* __restrict__ rate_W,     // [8]
    const float* __restrict__ rate_b,     // [1]
    const float* __restrict__ conv1_w,    // [4, 8, 3]
    const float* __restrict__ conv1_b,    // [4]
    const float* __restrict__ conv2_w,    // [4]
    const float* __restrict__ conv2_b,    // [1]
    const float* __restrict__ fusion,     // [2]
    float* __restrict__ out)              // [B, D]
{
    // LUT: for each (p, 3-bit spike window pat, c) the conv1 contribution.
    // Stored as float4 over c -> single ds_load_b128 per (p, t).
    __shared__ float4 lut4[64];           // [p][pat] -> (c0,c1,c2,c3)
    const int tid = threadIdx.x;
    {
        const int p = tid >> 5, pat = (tid >> 2) & 7, c = tid & 3;
        const float* w = conv1_w + (c * P_DIM + p) * 3;
        ((float*)lut4)[tid] = w[0] * (float)(pat & 1)
                            + w[1] * (float)((pat >> 1) & 1)
                            + w[2] * (float)((pat >> 2) & 1);
    }
    __syncthreads();

    const int g = blockIdx.x * blockDim.x + tid;   // g = b*D + d
    const float* Ib = I + (size_t)g * P_DIM;

    float Iv[8], mem[8], thr[8];
    uint32_t mask[8];
#pragma unroll
    for (int p = 0; p < 8; ++p) {
        Iv[p]  = Ib[p];
        thr[p] = thresholds[p];
        mem[p] = 0.0f;
        mask[p] = 0u;
    }

    // LIF: reset = H(mem-thr); mem = beta*mem + I - reset*thr; spk = H(mem-thr)
    for (int t = 0; t < NSTEPS; ++t) {
#pragma unroll
        for (int p = 0; p < 8; ++p) {
            const float r = (mem[p] > thr[p]) ? thr[p] : 0.0f;
            mem[p] = BETA * mem[p] + Iv[p] - r;
            mask[p] |= (mem[p] > thr[p] ? 1u : 0u) << t;
        }
    }

    // --- rate branch: rates = popcount/32 ---
    float rate_dec = rate_b[0];
#pragma unroll
    for (int p = 0; p < 8; ++p)
        rate_dec += (float)__popc((int)mask[p]) * (1.0f / 32.0f) * rate_W[p];

    // --- temporal branch: conv1(K=3, SAME) + relu, conv2(1x1), mean over T ---
    uint64_t m64[8];
#pragma unroll
    for (int p = 0; p < 8; ++p) m64[p] = (uint64_t)mask[p] << 1;  // zero-pad t=-1

    const float b10 = conv1_b[0], b11 = conv1_b[1];
    const float b12 = conv1_b[2], b13 = conv1_b[3];
    float hs0 = 0.f, hs1 = 0.f, hs2 = 0.f, hs3 = 0.f;
    for (int t = 0; t < NSTEPS; ++t) {
        float a0 = b10, a1 = b11, a2 = b12, a3 = b13;
#pragma unroll
        for (int p = 0; p < 8; ++p) {
            const int pat = (int)((m64[p] >> t) & 7u);   // s(t-1)|s(t)<<1|s(t+1)<<2
            const float4 l = lut4[(p << 3) | pat];
            a0 += l.x; a1 += l.y; a2 += l.z; a3 += l.w;
        }
        hs0 += fmaxf(a0, 0.f); hs1 += fmaxf(a1, 0.f);
        hs2 += fmaxf(a2, 0.f); hs3 += fmaxf(a3, 0.f);
    }
    const float temp_dec = conv2_b[0] +
        (conv2_w[0] * hs0 + conv2_w[1] * hs1 +
         conv2_w[2] * hs2 + conv2_w[3] * hs3) * (1.0f / 32.0f);

    // --- fusion: softmax([0.6, 0.4]) ---
    const float e0 = __expf(fusion[0]);
    const float e1 = __expf(fusion[1]);
    const float fw0 = e0 / (e0 + e1);
    out[g] = fw0 * rate_dec + (1.0f - fw0) * temp_dec;
}

// ---------------------------------------------------------------------------
extern "C" void kernel_launch(void* const* d_in, const int* in_sizes, int n_in,
                              void* d_out, int out_size, void* d_ws, size_t ws_size,
                              hipStream_t stream) {
    (void)in_sizes; (void)n_in; (void)out_size; (void)ws_size;
    const float* x      = (const float*)d_in[0];
    const float* W_proj = (const float*)d_in[1];
    const float* b_proj = (const float*)d_in[2];
    const float* thr    = (const float*)d_in[3];
    const float* rate_W = (const float*)d_in[4];
    const float* rate_b = (const float*)d_in[5];
    const float* c1w    = (const float*)d_in[6];
    const float* c1b    = (const float*)d_in[7];
    const float* c2w    = (const float*)d_in[8];
    const float* c2b    = (const float*)d_in[9];
    const float* fusion = (const float*)d_in[10];

    float* I = (float*)d_ws;                 // [1024, 2048] fp32 = 8 MB scratch

    // GEMM: 64 M-tiles x 32 N-supertiles = 2048 waves, 8 waves/block
    const int waves = (B_DIM / 16) * (TOT / 64);
    pc_gemm_wmma<<<waves / 8, 256, 0, stream>>>(x, W_proj, b_proj, I);

    // Fused SNN: one thread per (b, d) -> 262144 threads
    pc_fused<<<(B_DIM * D_DIM) / 256, 256, 0, stream>>>(
        I, thr, rate_W, rate_b, c1w, c1b, c2w, c2b, fusion, (float*)d_out);
}